// GlobalModel_69337952026908
// MI455X (gfx1250) — compile-verified
//
#include <hip/hip_runtime.h>
#include <hip/hip_bf16.h>

#define NODE_DIM   128
#define EDGE_DIM   32
#define GLOBAL_DIM 64
#define HIDDEN_DIM 256
#define IN_DIM     224   // 128 + 32 + 64
#define NSEG_MAX   512

typedef float v2f __attribute__((ext_vector_type(2)));
typedef float v8f __attribute__((ext_vector_type(8)));

__device__ __forceinline__ void gAtomicAddF(float* p, float v) {
    __hip_atomic_fetch_add(p, v, __ATOMIC_RELAXED, __HIP_MEMORY_SCOPE_AGENT);
}

// ---------------------------------------------------------------- zero ws
__global__ void zero_f32(float* __restrict__ p, int n) {
    int i = blockIdx.x * blockDim.x + threadIdx.x;
    if (i < n) p[i] = 0.0f;
}

// ------------------------------------------------- node segment reduction
// batch is sorted: each block (128 threads = feature lanes) walks 64
// consecutive nodes, accumulating per-segment partial sums in registers and
// flushing one atomic per (segment, feature) boundary crossing.
#define NODES_PER_BLOCK 64
__global__ void node_scatter(const float* __restrict__ x,
                             const int* __restrict__ batch,
                             float* __restrict__ node_sum,
                             float* __restrict__ node_cnt,
                             int n_nodes) {
    int f = threadIdx.x;                       // 0..127
    int start = blockIdx.x * NODES_PER_BLOCK;
    if (start >= n_nodes) return;
    int end = min(start + NODES_PER_BLOCK, n_nodes);

    int curseg = batch[start];
    float acc = 0.0f, cacc = 0.0f;
    for (int n = start; n < end; ++n) {
        int seg = batch[n];                    // uniform across block -> cached
        if (seg != curseg) {
            gAtomicAddF(&node_sum[(size_t)curseg * NODE_DIM + f], acc);
            if (f == 0) gAtomicAddF(&node_cnt[curseg], cacc);
            acc = 0.0f; cacc = 0.0f; curseg = seg;
        }
        acc  += x[(size_t)n * NODE_DIM + f];   // coalesced 512B per block row
        cacc += 1.0f;
    }
    gAtomicAddF(&node_sum[(size_t)curseg * NODE_DIM + f], acc);
    if (f == 0) gAtomicAddF(&node_cnt[curseg], cacc);
}

// ------------------------------------------------- edge segment reduction
// Random segment ids: stage the whole [512 x 32] accumulator in LDS (64KB of
// the WGP's 320KB). One wave per edge, lane == feature (wave32 == EDGE_DIM).
__global__ void edge_scatter(const float* __restrict__ edge_attr,
                             const int* __restrict__ edge_src,   // edge_index row 0
                             const int* __restrict__ batch,
                             float* __restrict__ edge_sum,
                             float* __restrict__ edge_cnt,
                             int n_edges, int nseg) {
    __shared__ float lsum[NSEG_MAX * EDGE_DIM];  // 64 KB
    __shared__ float lcnt[NSEG_MAX];             // 2 KB
    for (int i = threadIdx.x; i < NSEG_MAX * EDGE_DIM; i += blockDim.x) lsum[i] = 0.0f;
    for (int i = threadIdx.x; i < NSEG_MAX; i += blockDim.x) lcnt[i] = 0.0f;
    __syncthreads();

    int lane = threadIdx.x & 31;
    int wave = threadIdx.x >> 5;
    int wavesPerBlock = blockDim.x >> 5;
    int gwave = blockIdx.x * wavesPerBlock + wave;
    int nwaves = gridDim.x * wavesPerBlock;

    for (int e = gwave; e < n_edges; e += nwaves) {
        int seg = batch[edge_src[e]];          // broadcast load (same addr all lanes)
        float v = edge_attr[(size_t)e * EDGE_DIM + lane];  // coalesced 128B
        atomicAdd(&lsum[seg * EDGE_DIM + lane], v);        // ds_add_f32
        if (lane == 0) atomicAdd(&lcnt[seg], 1.0f);
    }
    __syncthreads();

    for (int i = threadIdx.x; i < nseg * EDGE_DIM; i += blockDim.x) {
        float v = lsum[i];
        if (v != 0.0f) gAtomicAddF(&edge_sum[i], v);
    }
    for (int i = threadIdx.x; i < nseg; i += blockDim.x) {
        float v = lcnt[i];
        if (v != 0.0f) gAtomicAddF(&edge_cnt[i], v);
    }
}

// ------------------------------------------------------- assemble H0 input
__global__ void assemble(const float* __restrict__ node_sum,
                         const float* __restrict__ node_cnt,
                         const float* __restrict__ edge_sum,
                         const float* __restrict__ edge_cnt,
                         const float* __restrict__ u,
                         float* __restrict__ H0) {
    int b = blockIdx.x;
    int j = threadIdx.x;   // 0..223
    float v;
    if (j < NODE_DIM) {
        v = node_sum[(size_t)b * NODE_DIM + j] / fmaxf(node_cnt[b], 1.0f);
    } else if (j < NODE_DIM + EDGE_DIM) {
        v = edge_sum[(size_t)b * EDGE_DIM + (j - NODE_DIM)] / fmaxf(edge_cnt[b], 1.0f);
    } else {
        v = u[(size_t)b * GLOBAL_DIM + (j - NODE_DIM - EDGE_DIM)];
    }
    H0[(size_t)b * IN_DIM + j] = v;
}

// ------------------------------------------------------------ WMMA GEMM
// C[M,N] = act(A[M,K] @ W[K,N] + bias[N]) using V_WMMA_F32_16X16X4_F32.
// One wave per 16x16 output tile. Per ISA VGPR layouts:
//   A 16x4 : lane row = lane&15; VGPR0/1 hold K = k0+2*half, k0+2*half+1
//   B 4x16 : lane col = lane&15; VGPR0/1 hold rows K = k0+2*half, k0+2*half+1
//   C/D    : VGPR v -> M = v + 8*half, N = lane&15
template <bool RELU>
__global__ void gemm_wmma(const float* __restrict__ A,
                          const float* __restrict__ W,
                          const float* __restrict__ bias,
                          float* __restrict__ C,
                          int M, int N, int K) {
    int lane = threadIdx.x & 31;
    int wave = threadIdx.x >> 5;
    int wavesPerBlock = blockDim.x >> 5;
    int tile = blockIdx.x * wavesPerBlock + wave;
    int ntilesN = N >> 4;
    int tileM = tile / ntilesN;
    int tileN = tile - tileM * ntilesN;
    if (tileM >= (M >> 4)) return;     // wave-uniform exit: EXEC stays all-ones

    int half = lane >> 4;              // 0 or 1
    int l15  = lane & 15;
    int row  = (tileM << 4) + l15;
    int col  = (tileN << 4) + l15;

    const float* __restrict__ Arow = A + (size_t)row * K;
    v8f acc = {};
    for (int k0 = 0; k0 < K; k0 += 4) {
        int ka = k0 + (half << 1);
        float2 av = *reinterpret_cast<const float2*>(Arow + ka);  // 8B aligned (ka even)
        v2f a; a.x = av.x; a.y = av.y;
        v2f b;
        b.x = W[(size_t)ka * N + col];
        b.y = W[(size_t)(ka + 1) * N + col];
        acc = __builtin_amdgcn_wmma_f32_16x16x4_f32(
            /*neg_a=*/false, a, /*neg_b=*/false, b,
            /*c_mod=*/(short)0, acc, /*reuse_a=*/false, /*reuse_b=*/false);
    }

    float bv = bias[col];
#pragma unroll
    for (int v = 0; v < 8; ++v) {
        int orow = (tileM << 4) + v + (half << 3);
        float r = acc[v] + bv;
        if (RELU) r = fmaxf(r, 0.0f);
        C[(size_t)orow * N + col] = r;
    }
}

// ----------------------------------------------------------------- launch
extern "C" void kernel_launch(void* const* d_in, const int* in_sizes, int n_in,
                              void* d_out, int out_size, void* d_ws, size_t ws_size,
                              hipStream_t stream) {
    const float* x         = (const float*)d_in[0];
    const int*   edge_idx  = (const int*)d_in[1];   // [2, E] flat; row 0 = src
    const float* edge_attr = (const float*)d_in[2];
    const float* u         = (const float*)d_in[3];
    const int*   batch     = (const int*)d_in[4];
    const float* W1        = (const float*)d_in[5];
    const float* b1        = (const float*)d_in[6];
    const float* W2        = (const float*)d_in[7];
    const float* b2        = (const float*)d_in[8];
    const float* W3        = (const float*)d_in[9];
    const float* b3        = (const float*)d_in[10];
    float* out = (float*)d_out;

    const int n_nodes = in_sizes[0] / NODE_DIM;           // 100000
    const int n_edges = in_sizes[2] / EDGE_DIM;           // 1600000
    const int B       = in_sizes[3] / GLOBAL_DIM;         // 512

    // workspace layout (floats)
    float* ws = (float*)d_ws;
    float* node_sum = ws;                                  // 512*128
    float* node_cnt = node_sum + (size_t)NSEG_MAX * NODE_DIM;   // 512
    float* edge_sum = node_cnt + NSEG_MAX;                 // 512*32
    float* edge_cnt = edge_sum + (size_t)NSEG_MAX * EDGE_DIM;   // 512
    float* H0 = edge_cnt + NSEG_MAX;                       // 512*224
    float* H1 = H0 + (size_t)NSEG_MAX * IN_DIM;            // 512*256
    float* H2 = H1 + (size_t)NSEG_MAX * HIDDEN_DIM;        // 512*256

    const int n_acc = NSEG_MAX * NODE_DIM + NSEG_MAX + NSEG_MAX * EDGE_DIM + NSEG_MAX;
    zero_f32<<<(n_acc + 255) / 256, 256, 0, stream>>>(ws, n_acc);

    node_scatter<<<(n_nodes + NODES_PER_BLOCK - 1) / NODES_PER_BLOCK, NODE_DIM, 0, stream>>>(
        x, batch, node_sum, node_cnt, n_nodes);

    edge_scatter<<<240, 256, 0, stream>>>(
        edge_attr, edge_idx, batch, edge_sum, edge_cnt, n_edges, B);

    assemble<<<B, IN_DIM, 0, stream>>>(node_sum, node_cnt, edge_sum, edge_cnt, u, H0);

    // Layer 1: [512,224] @ [224,256] -> relu -> H1
    {
        int tiles = (B / 16) * (HIDDEN_DIM / 16);
        gemm_wmma<true><<<(tiles + 7) / 8, 256, 0, stream>>>(H0, W1, b1, H1, B, HIDDEN_DIM, IN_DIM);
    }
    // Layer 2: [512,256] @ [256,256] -> relu -> H2
    {
        int tiles = (B / 16) * (HIDDEN_DIM / 16);
        gemm_wmma<true><<<(tiles + 7) / 8, 256, 0, stream>>>(H1, W2, b2, H2, B, HIDDEN_DIM, HIDDEN_DIM);
    }
    // Layer 3: [512,256] @ [256,64] -> out
    {
        int tiles = (B / 16) * (GLOBAL_DIM / 16);
        gemm_wmma<false><<<(tiles + 7) / 8, 256, 0, stream>>>(H2, W3, b3, out, B, GLOBAL_DIM, HIDDEN_DIM);
    }
    (void)n_in; (void)out_size; (void)ws_size;
}